// DynamicMarginRankingLoss_66666482368648
// MI455X (gfx1250) — compile-verified
//
#include <hip/hip_runtime.h>
#include <hip/hip_bf16.h>

typedef __attribute__((ext_vector_type(2))) float v2f;
typedef __attribute__((ext_vector_type(8))) float v8f;

#define TPB 256
#define JSLICES 8

// Reduce 64 per-wave partials (2 per lane: a0 -> A[m][0 or 2], a1 -> A[m][1 or 3])
// to a scalar broadcast to all lanes, using 5 chained V_WMMA_F32_16X16X4_F32.
// Step 1: D1 = A_data x B_ones          -> D1[i][j] = rowsum_i (16 row sums).
// Step 2: four WMMAs with A = (D1 vgpr pairs), B = ones, C chained
//         -> every element of final D = sum of all 16 row sums = total.
__device__ __forceinline__ float wmma_reduce64(float a0, float a1) {
  v2f a;    a.x = a0;  a.y = a1;
  v2f ones; ones.x = 1.0f; ones.y = 1.0f;
  v8f zero = {};
  v8f r = __builtin_amdgcn_wmma_f32_16x16x4_f32(false, a, false, ones, (short)0, zero, false, false);
  v2f p;
  v8f d = zero;
  p.x = r[0]; p.y = r[1];
  d = __builtin_amdgcn_wmma_f32_16x16x4_f32(false, p, false, ones, (short)0, d, false, false);
  p.x = r[2]; p.y = r[3];
  d = __builtin_amdgcn_wmma_f32_16x16x4_f32(false, p, false, ones, (short)0, d, false, false);
  p.x = r[4]; p.y = r[5];
  d = __builtin_amdgcn_wmma_f32_16x16x4_f32(false, p, false, ones, (short)0, d, false, false);
  p.x = r[6]; p.y = r[7];
  d = __builtin_amdgcn_wmma_f32_16x16x4_f32(false, p, false, ones, (short)0, d, false, false);
  return d[0];
}

// Kernel 1: margin = 0.25 * std(targets, ddof=1), two-pass for accuracy.
__global__ void dmrl_stats_kernel(const float* __restrict__ targets, float* __restrict__ ws, int n) {
  __shared__ float sred[TPB];
  const int tid = threadIdx.x;

  float s = 0.0f;
  for (int k = tid; k < n; k += TPB) s += targets[k];
  sred[tid] = s;
  __syncthreads();
  for (int off = TPB / 2; off > 0; off >>= 1) {
    if (tid < off) sred[tid] += sred[tid + off];
    __syncthreads();
  }
  const float mean = sred[0] / (float)n;
  __syncthreads();

  float ss = 0.0f;
  for (int k = tid; k < n; k += TPB) {
    float dlt = targets[k] - mean;
    ss += dlt * dlt;
  }
  sred[tid] = ss;
  __syncthreads();
  for (int off = TPB / 2; off > 0; off >>= 1) {
    if (tid < off) sred[tid] += sred[tid + off];
    __syncthreads();
  }
  if (tid == 0) {
    float var = sred[0] / (float)(n - 1);
    ws[0] = 0.25f * sqrtf(var);
  }
}

// Kernel 2: full N x N loss sum (symmetric; fixed up in finalize).
// Each block: i-tile (blockIdx.x) x j-slice (blockIdx.y). j-slice staged in LDS
// interleaved (t,p) as float2 so the inner loop is one ds_load_b64 broadcast.
__global__ void dmrl_pair_kernel(const float* __restrict__ preds,
                                 const float* __restrict__ targets,
                                 const float* __restrict__ ws,
                                 float* __restrict__ partials,
                                 int n, int chunk) {
  extern __shared__ float smem[];
  float2* tile = (float2*)smem;           // chunk entries of (t_j, p_j)
  float*  red  = smem + 2 * chunk;        // TPB partials

  const int tid = threadIdx.x;
  const int j0  = blockIdx.y * chunk;
  int j1 = j0 + chunk; if (j1 > n) j1 = n;
  const int cnt = j1 - j0;

  for (int k = tid; k < cnt; k += TPB) {
    float2 v; v.x = targets[j0 + k]; v.y = preds[j0 + k];
    tile[k] = v;
  }
  const float margin = ws[0];
  __syncthreads();

  const int i = blockIdx.x * TPB + tid;
  float acc0 = 0.0f, acc1 = 0.0f, acc2 = 0.0f, acc3 = 0.0f;
  if (i < n) {
    const float ti = targets[i];
    const float pi = preds[i];
    const int cnt4 = cnt & ~3;
    for (int jj = 0; jj < cnt4; jj += 4) {
      float2 e0 = tile[jj + 0];
      float2 e1 = tile[jj + 1];
      float2 e2 = tile[jj + 2];
      float2 e3 = tile[jj + 3];
      {
        float td = ti - e0.x, pd = e0.y - pi;
        float v = (td > 0.0f) ? pd : ((td < 0.0f) ? -pd : 0.0f);
        acc0 += fmaxf(v + margin, 0.0f);
      }
      {
        float td = ti - e1.x, pd = e1.y - pi;
        float v = (td > 0.0f) ? pd : ((td < 0.0f) ? -pd : 0.0f);
        acc1 += fmaxf(v + margin, 0.0f);
      }
      {
        float td = ti - e2.x, pd = e2.y - pi;
        float v = (td > 0.0f) ? pd : ((td < 0.0f) ? -pd : 0.0f);
        acc2 += fmaxf(v + margin, 0.0f);
      }
      {
        float td = ti - e3.x, pd = e3.y - pi;
        float v = (td > 0.0f) ? pd : ((td < 0.0f) ? -pd : 0.0f);
        acc3 += fmaxf(v + margin, 0.0f);
      }
    }
    for (int jj = cnt4; jj < cnt; ++jj) {
      float2 e = tile[jj];
      float td = ti - e.x, pd = e.y - pi;
      float v = (td > 0.0f) ? pd : ((td < 0.0f) ? -pd : 0.0f);
      acc0 += fmaxf(v + margin, 0.0f);
    }
  }
  red[tid] = (acc0 + acc1) + (acc2 + acc3);
  __syncthreads();

  // Wave 0 only (uniform branch -> EXEC all-ones inside the wave, as WMMA requires).
  if (tid < 32) {
    const float* r8 = red + tid * 8;
    float a0 = (r8[0] + r8[1]) + (r8[2] + r8[3]);
    float a1 = (r8[4] + r8[5]) + (r8[6] + r8[7]);
    float tot = wmma_reduce64(a0, a1);   // broadcast to all lanes
    if (tid == 0) partials[blockIdx.x * gridDim.y + blockIdx.y] = tot;
  }
}

// Kernel 3: ordered (deterministic) sum of partials + symmetry/diagonal fixup.
__global__ void dmrl_finalize_kernel(const float* __restrict__ ws,
                                     float* __restrict__ out,
                                     int n, int nparts) {
  if (threadIdx.x == 0 && blockIdx.x == 0) {
    float margin = ws[0];
    float s = 0.0f;
    for (int b = 0; b < nparts; ++b) s += ws[1 + b];
    // full-matrix sum S = 2*S_strict + n*margin  (diagonal: relu(margin) = margin)
    double strict = ((double)s - (double)n * (double)margin) * 0.5;
    double num_pairs = (double)n * (double)(n - 1) * 0.5;
    out[0] = (float)(strict / num_pairs);
  }
}

extern "C" void kernel_launch(void* const* d_in, const int* in_sizes, int n_in,
                              void* d_out, int out_size, void* d_ws, size_t ws_size,
                              hipStream_t stream) {
  const float* preds   = (const float*)d_in[0];
  const float* targets = (const float*)d_in[1];
  float* out = (float*)d_out;
  float* ws  = (float*)d_ws;      // ws[0]=margin, ws[1..nparts]=block partials
  const int n = in_sizes[0];

  dmrl_stats_kernel<<<1, TPB, 0, stream>>>(targets, ws, n);

  const int iblocks = (n + TPB - 1) / TPB;
  const int chunk   = (n + JSLICES - 1) / JSLICES;
  dim3 grid(iblocks, JSLICES);
  size_t shmem = (size_t)(2 * chunk + TPB) * sizeof(float);
  dmrl_pair_kernel<<<grid, TPB, shmem, stream>>>(preds, targets, ws, ws + 1, n, chunk);

  dmrl_finalize_kernel<<<1, 1, 0, stream>>>(ws, out, n, iblocks * JSLICES);
}